// GCNWithLinear_91216515432582
// MI455X (gfx1250) — compile-verified
//
#include <hip/hip_runtime.h>
#include <hip/hip_bf16.h>

// ---------------------------------------------------------------------------
// GCN: h = relu(bn(feat@W_in+b)); 3x { agg = D^-1/2 A D^-1/2 h; h = relu(bn(agg@W+b)) };
//      out = h@W_out + b_out
// N=100000, E=1.6M, H=128, C=47. Memory-bound on the edge scatter (~5GB ->
// ~215us @ 23.3TB/s). GEMMs: CDNA5 V_WMMA_F32_16X16X4_F32 (full fp32), A-slab
// staged to LDS by the Tensor Data Mover (TENSOR_LOAD_TO_LDS) with HW padding
// for bank-conflict-free fragment reads and zero-filled OOB rows.
// ---------------------------------------------------------------------------

typedef float v2f __attribute__((ext_vector_type(2)));
typedef float v8f __attribute__((ext_vector_type(8)));
typedef unsigned int u32x4 __attribute__((ext_vector_type(4)));
typedef int i32x4 __attribute__((ext_vector_type(4)));
typedef int i32x8 __attribute__((ext_vector_type(8)));

#define HDIM 128
#define TILE_ROWS 128
#define LDA 132   // 128 data + 4 pad floats per row (TDM pad_interval=128dw, pad_amount=4dw)

#if __has_builtin(__builtin_amdgcn_tensor_load_to_lds)
#define USE_TDM 1
#else
#define USE_TDM 0
#endif

// ------------------------------ utility kernels -----------------------------

__global__ void zero4_kernel(float4* __restrict__ p, long n4) {
  long i = (long)blockIdx.x * blockDim.x + threadIdx.x;
  if (i < n4) p[i] = make_float4(0.f, 0.f, 0.f, 0.f);
}

__global__ void degree_kernel(const int* __restrict__ src, const int* __restrict__ dst,
                              unsigned* __restrict__ deg_out, unsigned* __restrict__ deg_in,
                              int E) {
  int e = blockIdx.x * blockDim.x + threadIdx.x;
  if (e < E) {
    atomicAdd(&deg_out[src[e]], 1u);
    atomicAdd(&deg_in[dst[e]], 1u);
  }
}

__global__ void norm_kernel(const unsigned* __restrict__ deg, float* __restrict__ nrm, int N) {
  int i = blockIdx.x * blockDim.x + threadIdx.x;
  if (i < N) nrm[i] = rsqrtf(fmaxf((float)deg[i], 1.0f));
}

// Fold bias + eval-mode BatchNorm into per-column affine: y = x*a + b.
__global__ void prep_kernel(const float* __restrict__ b_in, const float* __restrict__ bc,
                            const float* __restrict__ gamma, const float* __restrict__ beta,
                            const float* __restrict__ rmean, const float* __restrict__ rvar,
                            const float* __restrict__ b_out,
                            float* __restrict__ colA, float* __restrict__ colB) {
  int j = threadIdx.x;          // 0..127
  int li = blockIdx.x;          // 0..4
  if (li < 4) {
    float a = gamma[li * HDIM + j] * rsqrtf(rvar[li * HDIM + j] + 1e-5f);
    float bias = (li == 0) ? b_in[j] : bc[(li - 1) * HDIM + j];
    colA[li * HDIM + j] = a;
    colB[li * HDIM + j] = beta[li * HDIM + j] + (bias - rmean[li * HDIM + j]) * a;
  } else {
    colA[4 * HDIM + j] = 1.0f;
    colB[4 * HDIM + j] = (j < 47) ? b_out[j] : 0.0f;
  }
}

// --------------------------------- scatter ----------------------------------
// One wave per edge: agg[dst] += h[src] * norm_src[src] * norm_dst[dst]
// (norm_dst folded here so the downstream GEMM needs no row scaling)
__global__ void __launch_bounds__(256) scatter_kernel(
    const float* __restrict__ h, const int* __restrict__ src, const int* __restrict__ dst,
    const float* __restrict__ norm_src, const float* __restrict__ norm_dst,
    float* __restrict__ agg, int E) {
  long gtid = (long)blockIdx.x * blockDim.x + threadIdx.x;
  long e = gtid >> 5;
  int lane = threadIdx.x & 31;
  if (e >= E) return;
  int s = src[e];
  int d = dst[e];
  float w = norm_src[s] * norm_dst[d];
  const float* hp = h + (size_t)s * HDIM;
  float* ap = agg + (size_t)d * HDIM;
#pragma unroll
  for (int i = 0; i < 4; ++i) {
    int c = lane + i * 32;
    unsafeAtomicAdd(&ap[c], hp[c] * w);   // global_atomic_add_f32
  }
}

// ----------------------------------- GEMM -----------------------------------
// Out[nrows, ncols] = epilogue( A[nrows,128] @ B[128,ncols] ),
// epilogue: y = x*colA[j] + colB[j], optional relu.
// Block: 256 thr = 8 waves over 128 rows; wave ct owns 16 cols (8 row tiles,
// 64 accumulator VGPRs). K loop: 32 steps of V_WMMA_F32_16X16X4_F32.
// A slab staged to LDS by TDM (zero-fill for rows >= nrows, 4-dword row pad).
__global__ void __launch_bounds__(256) gemm_bn_relu_kernel(
    const float* __restrict__ A, const float* __restrict__ B,
    const float* __restrict__ colA, const float* __restrict__ colB,
    float* __restrict__ Out, int nrows, int ncols, int do_relu) {
  __shared__ float As[TILE_ROWS * LDA];

  int row0 = blockIdx.x * TILE_ROWS;
  int t = threadIdx.x;

#if USE_TDM
  // ---- stage A slab via Tensor Data Mover (one instruction, wave 0) ----
  if (t < 32) {
    unsigned lds_addr = (unsigned)(uintptr_t)&As[0];   // low 32 bits = LDS offset
    unsigned long long ga = (unsigned long long)(uintptr_t)(A + (size_t)row0 * HDIM);
    unsigned rows_rem = (unsigned)(nrows - row0);      // OOB rows -> zero-filled

    u32x4 g0;
    g0[0] = 1u;                                        // count=1 (user descriptor)
    g0[1] = lds_addr;                                  // lds_addr
    g0[2] = (unsigned)(ga & 0xFFFFFFFFull);            // global_addr[31:0]
    g0[3] = (unsigned)((ga >> 32) & 0x01FFFFFFull) | (2u << 30);  // addr[56:32] | type=2

    i32x8 g1;
    g1[0] = (2 << 16)      // data_size = 4B
          | (1 << 20)      // pad_enable
          | (6 << 22)      // pad_interval: every 128 dwords
          | (3 << 25);     // pad_amount: 4 dwords  -> LDS row stride 132 floats
    g1[1] = (int)((HDIM & 0xFFFF) << 16);                       // tensor_dim0 lo16
    g1[2] = (int)((HDIM >> 16) | ((rows_rem & 0xFFFF) << 16));  // dim0 hi | dim1 lo
    g1[3] = (int)(((rows_rem >> 16) & 0xFFFF) | (HDIM << 16));  // dim1 hi | tile_dim0=128
    g1[4] = TILE_ROWS;                                          // tile_dim1=128 (tile_dim2=0)
    g1[5] = HDIM;                                               // tensor_dim0_stride lo
    g1[6] = 0;                                                  // stride hi / dim1_stride lo
    g1[7] = 0;

    i32x4 gz = {0, 0, 0, 0};
#if __clang_major__ >= 23
    i32x8 gz8 = {0, 0, 0, 0, 0, 0, 0, 0};
    __builtin_amdgcn_tensor_load_to_lds(g0, g1, gz, gz, gz8, 0);
#else
    __builtin_amdgcn_tensor_load_to_lds(g0, g1, gz, gz, 0);
#endif
    __builtin_amdgcn_s_wait_tensorcnt(0);
  }
  __syncthreads();
#else
  // ---- fallback: manual cooperative staging ----
#pragma unroll
  for (int i = 0; i < 16; ++i) {
    int v = t + i * 256;
    int r = v >> 5;
    int c4 = v & 31;
    int gr = row0 + r;
    float4 val = make_float4(0.f, 0.f, 0.f, 0.f);
    if (gr < nrows) val = *(const float4*)(A + (size_t)gr * HDIM + c4 * 4);
    *(float4*)(&As[r * LDA + c4 * 4]) = val;
  }
  __syncthreads();
#endif

  int wave = t >> 5;
  int lane = t & 31;
  int half = lane >> 4;         // 0: K slots {0,1}; 1: K slots {2,3}
  int l15 = lane & 15;
  int ncol_tiles = (ncols + 15) >> 4;
  if (wave >= ncol_tiles) return;       // after barrier; inactive for ncols=47

  int col = wave * 16 + l15;
  bool colOK = col < ncols;
  int colLd = colOK ? col : (ncols - 1);   // clamped index: no divergence in K loop

  v8f acc[8];
#pragma unroll
  for (int rt = 0; rt < 8; ++rt) acc[rt] = (v8f){0.f,0.f,0.f,0.f,0.f,0.f,0.f,0.f};

  // ---- K loop: 128 = 32 x 4 ----
  for (int kk = 0; kk < 32; ++kk) {
    int kA = kk * 4 + half * 2;
    v2f b;                                   // B frag: lane-in-half = N, slots = K
    b.x = B[(size_t)kA * ncols + colLd];
    b.y = B[(size_t)(kA + 1) * ncols + colLd];
#pragma unroll
    for (int rt = 0; rt < 8; ++rt) {
      v2f a = *(const v2f*)(&As[(rt * 16 + l15) * LDA + kA]);  // A frag: lane = M
      acc[rt] = __builtin_amdgcn_wmma_f32_16x16x4_f32(
          false, a, false, b, (short)0, acc[rt], false, false);
    }
  }

  // ---- epilogue: fused bias+BN (+relu) ----
  float sa = colA[colLd];
  float sb = colB[colLd];
  if (colOK) {
    if (row0 + TILE_ROWS <= nrows) {
      // fast path: full 128-row block, unguarded stores
#pragma unroll
      for (int rt = 0; rt < 8; ++rt) {
#pragma unroll
        for (int v = 0; v < 8; ++v) {
          int gr = row0 + rt * 16 + v + half * 8;   // C/D layout: M = v (+8 hi half)
          float y = acc[rt][v] * sa + sb;
          if (do_relu) y = fmaxf(y, 0.f);
          Out[(size_t)gr * ncols + col] = y;
        }
      }
    } else {
#pragma unroll
      for (int rt = 0; rt < 8; ++rt) {
#pragma unroll
        for (int v = 0; v < 8; ++v) {
          int gr = row0 + rt * 16 + v + half * 8;
          if (gr < nrows) {
            float y = acc[rt][v] * sa + sb;
            if (do_relu) y = fmaxf(y, 0.f);
            Out[(size_t)gr * ncols + col] = y;
          }
        }
      }
    }
  }
}

// --------------------------------- launcher ---------------------------------

extern "C" void kernel_launch(void* const* d_in, const int* in_sizes, int n_in,
                              void* d_out, int out_size, void* d_ws, size_t ws_size,
                              hipStream_t stream) {
  const float* feat  = (const float*)d_in[0];
  const int*   src   = (const int*)d_in[1];
  const int*   dst   = (const int*)d_in[2];
  const float* W_in  = (const float*)d_in[3];
  const float* b_in  = (const float*)d_in[4];
  const float* Wc    = (const float*)d_in[5];
  const float* bc    = (const float*)d_in[6];
  const float* gamma = (const float*)d_in[7];
  const float* beta  = (const float*)d_in[8];
  const float* rmean = (const float*)d_in[9];
  const float* rvar  = (const float*)d_in[10];
  const float* W_out = (const float*)d_in[11];
  const float* b_out = (const float*)d_in[12];
  float* out = (float*)d_out;

  int N = in_sizes[0] / HDIM;
  int E = in_sizes[1];

  // workspace layout (256B aligned)
  char* ws = (char*)d_ws;
  size_t off = 0;
  auto alloc = [&](size_t bytes) -> void* {
    void* p = ws + off;
    off = (off + bytes + 255) & ~(size_t)255;
    return p;
  };
  float*    bufA     = (float*)alloc((size_t)N * HDIM * sizeof(float));
  float*    bufB     = (float*)alloc((size_t)N * HDIM * sizeof(float));
  float*    norm_src = (float*)alloc((size_t)N * sizeof(float));
  float*    norm_dst = (float*)alloc((size_t)N * sizeof(float));
  unsigned* degs     = (unsigned*)alloc((size_t)2 * N * sizeof(unsigned)); // out|in
  unsigned* deg_o    = degs;
  unsigned* deg_i    = degs + N;
  float*    colA     = (float*)alloc(5 * HDIM * sizeof(float));
  float*    colB     = (float*)alloc(5 * HDIM * sizeof(float));
  (void)ws_size; (void)n_in; (void)out_size;

  // 1) degrees -> norms
  {
    long n4 = (long)(((size_t)2 * N * sizeof(unsigned) + 15) / 16);
    zero4_kernel<<<(n4 + 255) / 256, 256, 0, stream>>>((float4*)degs, n4);
  }
  degree_kernel<<<(E + 255) / 256, 256, 0, stream>>>(src, dst, deg_o, deg_i, E);
  norm_kernel<<<(N + 255) / 256, 256, 0, stream>>>(deg_o, norm_src, N);
  norm_kernel<<<(N + 255) / 256, 256, 0, stream>>>(deg_i, norm_dst, N);

  // 2) fold bias+BN into per-column affine
  prep_kernel<<<5, HDIM, 0, stream>>>(b_in, bc, gamma, beta, rmean, rvar, b_out, colA, colB);

  int gblocks = (N + TILE_ROWS - 1) / TILE_ROWS;

  // 3) input GEMM: bufA = relu(bn0(feat @ W_in + b_in))
  gemm_bn_relu_kernel<<<gblocks, 256, 0, stream>>>(feat, W_in, colA, colB, bufA, N, HDIM, 1);

  // 4) three GraphConv layers
  for (int i = 0; i < 3; ++i) {
    long nb4 = (long)N * (HDIM / 4);
    zero4_kernel<<<(nb4 + 255) / 256, 256, 0, stream>>>((float4*)bufB, nb4);
    long sthreads = (long)E * 32;
    scatter_kernel<<<(sthreads + 255) / 256, 256, 0, stream>>>(bufA, src, dst, norm_src,
                                                               norm_dst, bufB, E);
    gemm_bn_relu_kernel<<<gblocks, 256, 0, stream>>>(bufB, Wc + (size_t)i * HDIM * HDIM,
                                                     colA + (i + 1) * HDIM,
                                                     colB + (i + 1) * HDIM, bufA, N, HDIM, 1);
  }

  // 5) output GEMM: out = bufA @ W_out + b_out  (no BN, no relu)
  gemm_bn_relu_kernel<<<gblocks, 256, 0, stream>>>(bufA, W_out, colA + 4 * HDIM,
                                                   colB + 4 * HDIM, out, N, 47, 0);
}